// AtomCollection_81492709474764
// MI455X (gfx1250) — compile-verified
//
#include <hip/hip_runtime.h>
#include <hip/hip_bf16.h>
#include <stdint.h>

// ---------------- problem constants ----------------
#define N_ATOMS   512
#define KERN      256
#define PAD       128
#define BATCH     8
#define T_LEN     32768
#define T_OUT     32769                 // T + 2*pad - K + 1
#define XPAD_LEN  (T_LEN + 2 * PAD)     // 33024
#define KEEP      512
#define PER_B     ((size_t)N_ATOMS * T_OUT)   // 16,777,728 elements per batch

// GEMM tiling: block = 8 waves, each wave owns a 16x32 tile (2 accumulator chains)
#define TILE_N    256                   // time columns per block (8 waves x 32)
#define LDA_PAD   260                   // 16-row A tile LDS stride (float), bank-conflict free

typedef float v2f __attribute__((ext_vector_type(2)));
typedef float v8f __attribute__((ext_vector_type(8)));

// order-preserving map: larger float <-> larger uint
__device__ __forceinline__ uint32_t f2ord(float f) {
    uint32_t u = __float_as_uint(f);
    return (u & 0x80000000u) ? ~u : (u | 0x80000000u);
}

// ---------------- 1) normalize atoms ----------------
__global__ void k_normalize_atoms(const float* __restrict__ atoms,
                                  float* __restrict__ a_norm) {
    __shared__ float red[256];
    const int a   = blockIdx.x;
    const int tid = threadIdx.x;
    float v = atoms[a * KERN + tid];
    red[tid] = v * v;
    __syncthreads();
    for (int s = 128; s > 0; s >>= 1) {
        if (tid < s) red[tid] += red[tid + s];
        __syncthreads();
    }
    float norm = sqrtf(red[0]) + 1e-12f;
    a_norm[a * KERN + tid] = v / norm;
}

// ---------------- 2) build padded signal ----------------
__global__ void k_build_xpad(const float* __restrict__ x,
                             float* __restrict__ xpad) {
    size_t i = (size_t)blockIdx.x * blockDim.x + threadIdx.x;
    size_t total = (size_t)BATCH * XPAD_LEN;
    if (i >= total) return;
    int b = (int)(i / XPAD_LEN);
    int t = (int)(i % XPAD_LEN);
    float v = 0.f;
    if (t >= PAD && t < PAD + T_LEN) v = x[(size_t)b * T_LEN + (t - PAD)];
    xpad[i] = v;
}

// ---------------- 3) correlation GEMM via V_WMMA_F32_16X16X4_F32 ----------------
// grid: (ceil(T_OUT/TILE_N), N_ATOMS/16, BATCH), block: 256 (8 waves)
// Each wave: 16(M) x 32(N) output tile = two independent 16x16 accumulator
// chains, so consecutive v_wmma issues have no C->C RAW dependency and the
// A fragment is reused across both chains (12B/lane LDS traffic per wmma).
__global__ void k_gemm_wmma(const float* __restrict__ a_norm,
                            const float* __restrict__ xpad,
                            float* __restrict__ full) {
    __shared__ float ldsA[16 * LDA_PAD];          // 16 atoms x 256 taps (padded)
    __shared__ float ldsX[TILE_N + KERN + 8];     // 512 window samples (+slack)

    const int tid  = threadIdx.x;
    const int wave = tid >> 5;
    const int lane = tid & 31;
    const int b    = blockIdx.z;
    const int a0   = blockIdx.y * 16;
    const int t0   = blockIdx.x * TILE_N;

    // stage A tile: 16x256 floats, coalesced
    for (int i = tid; i < 16 * KERN; i += 256) {
        int r = i >> 8, c = i & (KERN - 1);
        ldsA[r * LDA_PAD + c] = a_norm[(size_t)(a0 + r) * KERN + c];
    }
    // stage X window: xpad[b, t0 .. t0+511]
    const size_t xb = (size_t)b * XPAD_LEN;
    for (int i = tid; i < TILE_N + KERN; i += 256) {
        int g = t0 + i;
        ldsX[i] = (g < XPAD_LEN) ? xpad[xb + g] : 0.f;
    }
    __syncthreads();

    // ISA 16x4 f32 A layout: lanes 0-15 hold K=(k,k+1), lanes 16-31 K=(k+2,k+3)
    const int row  = lane & 15;       // A row (M)
    const int hi   = lane >> 4;       // selects K pair
    const int n0   = (wave << 5) + (lane & 15);   // local col of chain 0
    const int n1   = n0 + 16;                     // local col of chain 1

    v8f acc0 = {};
    v8f acc1 = {};
#pragma unroll 8
    for (int k = 0; k < KERN; k += 4) {
        const int kk = k + hi * 2;
        v2f afrag = *(const v2f*)&ldsA[row * LDA_PAD + kk];   // A[row][kk..kk+1]
        v2f b0, b1;
        b0.x = ldsX[n0 + kk];       // B[kk][n]   = Xwin[n+kk]
        b0.y = ldsX[n0 + kk + 1];   // B[kk+1][n]
        b1.x = ldsX[n1 + kk];
        b1.y = ldsX[n1 + kk + 1];
        acc0 = __builtin_amdgcn_wmma_f32_16x16x4_f32(
            false, afrag, false, b0, (short)0, acc0, false, false);
        acc1 = __builtin_amdgcn_wmma_f32_16x16x4_f32(
            false, afrag, false, b1, (short)0, acc1, false, false);
    }

    // C/D layout: VGPR r -> M = r + hi*8, N = lane&15
    const size_t obase = ((size_t)b * N_ATOMS + a0) * T_OUT;
    const int tcol0 = t0 + n0;
    const int tcol1 = t0 + n1;
    if (tcol0 < T_OUT) {
#pragma unroll
        for (int r = 0; r < 8; ++r) {
            int m = r + hi * 8;
            full[obase + (size_t)m * T_OUT + tcol0] = acc0[r];
        }
    }
    if (tcol1 < T_OUT) {
#pragma unroll
        for (int r = 0; r < 8; ++r) {
            int m = r + hi * 8;
            full[obase + (size_t)m * T_OUT + tcol1] = acc1[r];
        }
    }
}

// ---------------- 4) init select state + zero recon ----------------
__global__ void k_init(uint32_t* __restrict__ hist, uint32_t* __restrict__ prefix,
                       uint32_t* __restrict__ krem, uint32_t* __restrict__ tie_budget,
                       uint32_t* __restrict__ tie_count, uint32_t* __restrict__ nz_count,
                       float* __restrict__ recon) {
    size_t i = (size_t)blockIdx.x * blockDim.x + threadIdx.x;
    if (i < (size_t)BATCH * T_LEN) recon[i] = 0.f;
    if (i < BATCH * 256) hist[i] = 0;
    if (i < BATCH) {
        prefix[i] = 0; krem[i] = KEEP; tie_budget[i] = 0;
        tie_count[i] = 0; nz_count[i] = 0;
    }
}

// ---------------- 5) radix-select histogram pass ----------------
// grid: (2048, BATCH), block 256
__global__ void k_hist(const float* __restrict__ full,
                       const uint32_t* __restrict__ prefix,
                       uint32_t* __restrict__ hist, int shift) {
    __shared__ uint32_t h[256];
    const int tid = threadIdx.x;
    const int b   = blockIdx.y;
    h[tid] = 0;
    __syncthreads();
    const uint32_t pfx = prefix[b];
    const size_t base   = (size_t)b * PER_B;
    const size_t stride = (size_t)gridDim.x * 256;
    for (size_t i = (size_t)blockIdx.x * 256 + tid; i < PER_B; i += stride) {
        uint32_t u = f2ord(full[base + i]);
        bool match = (shift == 24) || (((u ^ pfx) >> (shift + 8)) == 0);
        if (match) atomicAdd(&h[(u >> shift) & 255u], 1u);
    }
    __syncthreads();
    if (h[tid]) atomicAdd(&hist[b * 256 + tid], h[tid]);
}

// ---------------- 6) scan: pick bin, extend prefix, reset hist ----------------
// grid: BATCH blocks, 256 threads
__global__ void k_scan(uint32_t* __restrict__ hist, uint32_t* __restrict__ prefix,
                       uint32_t* __restrict__ krem, uint32_t* __restrict__ tie_budget,
                       int shift, int last) {
    const int b = blockIdx.x;
    if (threadIdx.x == 0) {
        uint32_t k = krem[b], c = 0;
        int bin = 255;
        for (; bin > 0; --bin) {
            uint32_t hv = hist[b * 256 + bin];
            if (c + hv >= k) break;
            c += hv;
        }
        prefix[b] |= ((uint32_t)bin) << shift;
        krem[b] = k - c;
        if (last) tie_budget[b] = k - c;
    }
    __syncthreads();
    hist[b * 256 + threadIdx.x] = 0;
}

// ---------------- 7) sparsify (top-k apply) + compact nonzeros ----------------
// grid: (2048, BATCH), block 256
__global__ void k_sparsify(const float* __restrict__ full,
                           float* __restrict__ feature_map,
                           const uint32_t* __restrict__ prefix,
                           const uint32_t* __restrict__ tie_budget,
                           uint32_t* __restrict__ tie_count,
                           uint32_t* __restrict__ nz_count,
                           uint32_t* __restrict__ nz_idx,
                           float* __restrict__ nz_val) {
    const int b = blockIdx.y;
    const uint32_t thr    = prefix[b];
    const uint32_t budget = tie_budget[b];
    const size_t base   = (size_t)b * PER_B;
    const size_t stride = (size_t)gridDim.x * 256;
    for (size_t i = (size_t)blockIdx.x * 256 + threadIdx.x; i < PER_B; i += stride) {
        float f = full[base + i];
        uint32_t u = f2ord(f);
        bool keep = false;
        if (u > thr) keep = true;
        else if (u == thr) keep = (atomicAdd(&tie_count[b], 1u) < budget);
        float outv = 0.f;
        if (keep) {
            outv = f;
            uint32_t slot = atomicAdd(&nz_count[b], 1u);
            if (slot < KEEP) {
                nz_idx[b * KEEP + slot] = (uint32_t)i;
                nz_val[b * KEEP + slot] = f;
            }
        }
        feature_map[base + i] = outv;
    }
}

// ---------------- 8) sparse transposed conv (scatter-add) ----------------
// grid: BATCH blocks, 256 threads (one per kernel tap)
__global__ void k_recon(const float* __restrict__ a_norm,
                        const uint32_t* __restrict__ nz_count,
                        const uint32_t* __restrict__ nz_idx,
                        const float* __restrict__ nz_val,
                        float* __restrict__ recon) {
    const int b = blockIdx.x;
    const int k = threadIdx.x;             // tap 0..255
    uint32_t cnt = nz_count[b];
    if (cnt > KEEP) cnt = KEEP;
    for (uint32_t e = 0; e < cnt; ++e) {
        uint32_t idx = nz_idx[b * KEEP + e];
        float    v   = nz_val[b * KEEP + e];
        int a  = (int)(idx / T_OUT);
        int tp = (int)(idx % T_OUT);
        int t  = tp + k - PAD;             // recon[t] += fm[a,tp] * a_norm[a,k]
        if (t >= 0 && t < T_LEN)
            atomicAdd(&recon[(size_t)b * T_LEN + t], v * a_norm[a * KERN + k]);
    }
}

// ---------------- launcher ----------------
extern "C" void kernel_launch(void* const* d_in, const int* in_sizes, int n_in,
                              void* d_out, int out_size, void* d_ws, size_t ws_size,
                              hipStream_t stream) {
    const float* x     = (const float*)d_in[0];   // (8,1,32768)
    const float* atoms = (const float*)d_in[1];   // (512,1,256)

    // d_out = feature_map | recon | full (flat, return order)
    float* feature_map = (float*)d_out;
    float* recon       = feature_map + (size_t)BATCH * PER_B;
    float* full        = recon + (size_t)BATCH * T_LEN;

    // workspace carve-up
    char* w = (char*)d_ws;
    float*    a_norm     = (float*)w;                 w += (size_t)N_ATOMS * KERN * 4;
    float*    xpad       = (float*)w;                 w += (size_t)BATCH * XPAD_LEN * 4;
    uint32_t* hist       = (uint32_t*)w;              w += (size_t)BATCH * 256 * 4;
    uint32_t* prefix     = (uint32_t*)w;              w += BATCH * 4;
    uint32_t* krem       = (uint32_t*)w;              w += BATCH * 4;
    uint32_t* tie_budget = (uint32_t*)w;              w += BATCH * 4;
    uint32_t* tie_count  = (uint32_t*)w;              w += BATCH * 4;
    uint32_t* nz_count   = (uint32_t*)w;              w += BATCH * 4;
    uint32_t* nz_idx     = (uint32_t*)w;              w += (size_t)BATCH * KEEP * 4;
    float*    nz_val     = (float*)w;                 w += (size_t)BATCH * KEEP * 4;

    // 1) normalize atoms
    k_normalize_atoms<<<N_ATOMS, 256, 0, stream>>>(atoms, a_norm);
    // 2) padded signal
    {
        size_t total = (size_t)BATCH * XPAD_LEN;
        k_build_xpad<<<(unsigned)((total + 255) / 256), 256, 0, stream>>>(x, xpad);
    }
    // 3) WMMA correlation GEMM -> full
    {
        dim3 grid((T_OUT + TILE_N - 1) / TILE_N, N_ATOMS / 16, BATCH);
        k_gemm_wmma<<<grid, 256, 0, stream>>>(a_norm, xpad, full);
    }
    // 4) init state + zero recon
    {
        size_t total = (size_t)BATCH * T_LEN;
        k_init<<<(unsigned)((total + 255) / 256), 256, 0, stream>>>(
            hist, prefix, krem, tie_budget, tie_count, nz_count, recon);
    }
    // 5/6) exact top-512 threshold via 4x 8-bit radix select
    for (int pass = 0; pass < 4; ++pass) {
        int shift = 24 - pass * 8;
        k_hist<<<dim3(2048, BATCH), 256, 0, stream>>>(full, prefix, hist, shift);
        k_scan<<<BATCH, 256, 0, stream>>>(hist, prefix, krem, tie_budget,
                                          shift, pass == 3 ? 1 : 0);
    }
    // 7) feature_map + nonzero compaction
    k_sparsify<<<dim3(2048, BATCH), 256, 0, stream>>>(
        full, feature_map, prefix, tie_budget, tie_count, nz_count, nz_idx, nz_val);
    // 8) sparse transposed conv
    k_recon<<<BATCH, 256, 0, stream>>>(a_norm, nz_count, nz_idx, nz_val, recon);
}